// SymmetricMatrixRegressor_24017457119283
// MI455X (gfx1250) — compile-verified
//
#include <hip/hip_runtime.h>
#include <math.h>

// ---------------------------------------------------------------------------
// MACE-like equivariant GNN forward for MI455X (gfx1250, wave32).
// WMMA f32 16x16x4 used for: (a) per-edge radial einsum R[E,8] @ Wrad[8,96],
// (b) node scalar update GEMM B0[N,32] @ Wm0[32,32] with fused SiLU.
// Scatter into A[N,C,9] via global fp32 atomics; readout reduced via LDS
// ds_add_f32 then 9 global atomics per block.
// ---------------------------------------------------------------------------

typedef float v2f __attribute__((ext_vector_type(2)));
typedef float v8f __attribute__((ext_vector_type(8)));

#define N_NODES_ 4096
#define N_EDGES_ 65536
#define G_       4
#define C_       32
#define NB_      8
#define NZ_      3

__device__ __forceinline__ v8f wmma_f32_4(v2f a, v2f b, v8f c) {
  // D = A(16x4 f32) * B(4x16 f32) + C(16x16 f32)
  return __builtin_amdgcn_wmma_f32_16x16x4_f32(false, a, false, b, (short)0, c,
                                               false, false);
}

__device__ __forceinline__ void atomAddG(float* p, float v) {
  __hip_atomic_fetch_add(p, v, __ATOMIC_RELAXED, __HIP_MEMORY_SCOPE_AGENT);
}
__device__ __forceinline__ void atomAddL(float* p, float v) {
  __hip_atomic_fetch_add(p, v, __ATOMIC_RELAXED, __HIP_MEMORY_SCOPE_WORKGROUP);
}

// ---------------------------------------------------------------------------
__global__ void __launch_bounds__(256) zero_kernel(float* p, int n) {
  int i = blockIdx.x * blockDim.x + threadIdx.x;
  if (i < n) p[i] = 0.f;
}

// h[n,c] = sum_z node_attr[n,z] * Wz[z,c]
__global__ void __launch_bounds__(256)
node_embed_kernel(const float* __restrict__ na, const float* __restrict__ Wz,
                  float* __restrict__ h) {
  int i = blockIdx.x * 256 + threadIdx.x;   // i < N*C
  int n = i >> 5, c = i & 31;
  float acc = 0.f;
#pragma unroll
  for (int z = 0; z < NZ_; ++z) acc += na[n * NZ_ + z] * Wz[z * C_ + c];
  h[i] = acc;
}

// ---------------------------------------------------------------------------
// Edge kernel: per wave = 16 edges. w[16,96] = R[16,8] @ Wrad[8,96] via WMMA,
// then m = (w * h[src]) (x) Y_l and atomic scatter-add into A[dst, c, 0..8].
__global__ void __launch_bounds__(256)
edge_message_kernel(const float* __restrict__ rr, const float* __restrict__ vv,
                    const int* __restrict__ src, const int* __restrict__ dst,
                    const float* __restrict__ h, const float* __restrict__ Wrad,
                    float* __restrict__ A) {
  __shared__ float sW[NB_ * 96];       // Wrad [8][96], col = c*3 + l
  __shared__ float sY[8][16][8];       // per-wave: 16 edges x {Y1[3], Y2[5]}
  __shared__ int   sS[8][16];
  __shared__ int   sD[8][16];

  const int tid = threadIdx.x;
  for (int i = tid; i < NB_ * 96; i += 256) sW[i] = Wrad[i];

  const int wave = tid >> 5;
  const int lane = tid & 31;
  const int l15  = lane & 15;
  const int kh   = lane >> 4;                  // K-half (WMMA A/B layout)
  const int base = (blockIdx.x * 8 + wave) * 16;

  if (kh == 0) {  // lanes 0..15: spherical harmonics + indices for 16 edges
    const int e = base + l15;
    float vx = vv[3 * e + 0], vy = vv[3 * e + 1], vz = vv[3 * e + 2];
    float inv = 1.0f / (sqrtf(vx * vx + vy * vy + vz * vz) + 1e-9f);
    float ux = vx * inv, uy = vy * inv, uz = vz * inv;
    const float sq3 = 1.7320508075688772f;
    const float cy2 = 2.7386127875258306f;   // sqrt(15/2)
    const float sq2 = 1.4142135623730951f;
    sY[wave][l15][0] = sq3 * ux;
    sY[wave][l15][1] = sq3 * uy;
    sY[wave][l15][2] = sq3 * uz;
    sY[wave][l15][3] = cy2 * sq2 * ux * uy;
    sY[wave][l15][4] = cy2 * sq2 * uy * uz;
    sY[wave][l15][5] = cy2 * (2.f * uz * uz - ux * ux - uy * uy) * 0.4082482904638631f;
    sY[wave][l15][6] = cy2 * sq2 * ux * uz;
    sY[wave][l15][7] = cy2 * (ux * ux - uy * uy) * 0.7071067811865475f;
    sS[wave][l15] = src[e];
    sD[wave][l15] = dst[e];
  }
  __syncthreads();

  // Bessel radial basis for this lane's A-fragment K slots.
  // R[b] = sqrt(2/rc) * ((b+1)*pi/rc) * sinc((b+1)*pi*r/rc) , rc = 6
  const float rv  = rr[base + l15];
  const float PI_ = 3.14159265358979f;
  float Ra[2][2];
#pragma unroll
  for (int s = 0; s < 2; ++s) {
#pragma unroll
    for (int v = 0; v < 2; ++v) {
      int b = 4 * s + 2 * kh + v;
      float nf = (float)(b + 1);
      float x  = PI_ * rv * nf * (1.0f / 6.0f);
      float snc = (fabsf(x) < 1e-8f) ? 1.0f : (sinf(x) / x);
      Ra[s][v] = 0.5773502691896258f * (nf * PI_ * (1.0f / 6.0f)) * snc;
    }
  }

#pragma unroll
  for (int t = 0; t < 6; ++t) {                 // 6 tiles of 16 of the 96 cols
    v8f acc = {0.f, 0.f, 0.f, 0.f, 0.f, 0.f, 0.f, 0.f};
#pragma unroll
    for (int s = 0; s < 2; ++s) {               // K = 8 as two K=4 steps
      int k0 = 4 * s + 2 * kh;
      v2f af; af.x = Ra[s][0]; af.y = Ra[s][1];
      v2f bf;
      bf.x = sW[k0 * 96 + t * 16 + l15];
      bf.y = sW[(k0 + 1) * 96 + t * 16 + l15];
      acc = wmma_f32_4(af, bf, acc);
    }
    // D layout: lane holds col n = l15; VGPR j holds edge row (8*kh + j)
    const int col = t * 16 + l15;
    const int cch = col / 3;
    const int lc  = col - 3 * cch;
#pragma unroll
    for (int j = 0; j < 8; ++j) {
      const int mm = kh * 8 + j;
      const int sn = sS[wave][mm];
      const int dn = sD[wave][mm];
      const float mv = acc[j] * h[sn * C_ + cch];
      float* Ap = A + (size_t)dn * (C_ * 9) + cch * 9;
      if (lc == 0) {
        atomAddG(Ap + 0, mv);                           // * Y0 = 1
      } else if (lc == 1) {
        atomAddG(Ap + 1, mv * sY[wave][mm][0]);
        atomAddG(Ap + 2, mv * sY[wave][mm][1]);
        atomAddG(Ap + 3, mv * sY[wave][mm][2]);
      } else {
#pragma unroll
        for (int i5 = 0; i5 < 5; ++i5)
          atomAddG(Ap + 4 + i5, mv * sY[wave][mm][3 + i5]);
      }
    }
  }
}

// ---------------------------------------------------------------------------
// Per-node 3-body product (B0, B2 only; B1 unused by readout) + l2 readout,
// reduced to a single 3x3 per graph. Also emits B0 for the scalar update.
__global__ void __launch_bounds__(256)
node_product_readout_kernel(const float* __restrict__ A, const float* __restrict__ Wp,
                            const float* __restrict__ Wr0, const float* __restrict__ Wr2,
                            float* __restrict__ B0out, float* __restrict__ outM) {
  __shared__ float sred[9];
  const int tid = threadIdx.x;
  if (tid < 9) sred[tid] = 0.f;
  __syncthreads();

  const int n = blockIdx.x * 256 + tid;
  float s = 0.f, t0 = 0.f, t1 = 0.f, t2 = 0.f, t3 = 0.f, t4 = 0.f;
  const float* Ab = A + (size_t)n * (C_ * 9);
#pragma unroll 4
  for (int c = 0; c < C_; ++c) {
    const float* Ap = Ab + c * 9;
    float A0 = Ap[0];
    float ax = Ap[1], ay = Ap[2], az = Ap[3];
    float b0 = Ap[4], b1 = Ap[5], b2 = Ap[6], b3 = Ap[7], b4 = Ap[8];
    float d11 = ax * ax + ay * ay + az * az;
    float d22 = b0 * b0 + b1 * b1 + b2 * b2 + b3 * b3 + b4 * b4;
    float B0c = Wp[c] * A0 * A0 + Wp[C_ + c] * d11 + Wp[2 * C_ + c] * d22;
    float w5 = Wp[5 * C_ + c], w6 = Wp[6 * C_ + c];
    const float s2 = 1.4142135623730951f;
    float t5a = s2 * ax * ay;
    float t5b = s2 * ay * az;
    float t5c = (2.f * az * az - ax * ax - ay * ay) * 0.4082482904638631f;
    float t5d = s2 * ax * az;
    float t5e = (ax * ax - ay * ay) * 0.7071067811865475f;
    float wr2 = Wr2[c];
    s  += Wr0[c] * B0c;
    t0 += wr2 * (w5 * A0 * b0 + w6 * t5a);
    t1 += wr2 * (w5 * A0 * b1 + w6 * t5b);
    t2 += wr2 * (w5 * A0 * b2 + w6 * t5c);
    t3 += wr2 * (w5 * A0 * b3 + w6 * t5d);
    t4 += wr2 * (w5 * A0 * b4 + w6 * t5e);
    B0out[n * C_ + c] = B0c;
  }
  const float r2 = 0.7071067811865475f, r6 = 0.4082482904638631f,
              r3 = 0.5773502691896258f;
  float M00 = -t2 * r6 + t4 * r2 + s * r3;
  float M11 = -t2 * r6 - t4 * r2 + s * r3;
  float M22 = 2.f * t2 * r6 + s * r3;
  float M01 = t0 * r2, M12 = t1 * r2, M02 = t3 * r2;
  float Mv[9] = {M00, M01, M02, M01, M11, M12, M02, M12, M22};
#pragma unroll
  for (int i = 0; i < 9; ++i) atomAddL(&sred[i], Mv[i]);
  __syncthreads();
  if (tid < 9) atomAddG(outM + tid, sred[tid]);
}

// ---------------------------------------------------------------------------
// f0 = silu(B0[N,32] @ Wm0[32,32]) via WMMA f32 16x16x4, 8 chained K-steps.
__global__ void __launch_bounds__(256)
scalar_update_gemm_kernel(const float* __restrict__ B0, const float* __restrict__ Wm0,
                          float* __restrict__ f0) {
  const int tid  = threadIdx.x;
  const int wave = tid >> 5;
  const int lane = tid & 31;
  const int l15  = lane & 15;
  const int kh   = lane >> 4;
  const int tile = blockIdx.x * 8 + wave;     // (N/16) * 2 tiles
  const int mt = tile >> 1, nt = tile & 1;
  const int colB = nt * 16 + l15;
  const int rowA = mt * 16 + l15;
  v8f acc = {0.f, 0.f, 0.f, 0.f, 0.f, 0.f, 0.f, 0.f};
#pragma unroll
  for (int s = 0; s < 8; ++s) {
    const int k0 = 4 * s + 2 * kh;
    v2f af; af.x = B0[rowA * C_ + k0];  af.y = B0[rowA * C_ + k0 + 1];
    v2f bf; bf.x = Wm0[k0 * C_ + colB]; bf.y = Wm0[(k0 + 1) * C_ + colB];
    acc = wmma_f32_4(af, bf, acc);
  }
#pragma unroll
  for (int j = 0; j < 8; ++j) {
    float x = acc[j];
    f0[(mt * 16 + kh * 8 + j) * C_ + colB] = x / (1.0f + expf(-x));
  }
}

// ---------------------------------------------------------------------------
extern "C" void kernel_launch(void* const* d_in, const int* in_sizes, int n_in,
                              void* d_out, int out_size, void* d_ws, size_t ws_size,
                              hipStream_t stream) {
  (void)in_sizes; (void)n_in; (void)out_size;
  const float* x     = (const float*)d_in[0];   // [G,E] radii
  const float* xv    = (const float*)d_in[1];   // [G,E,3] edge vectors
  const float* na    = (const float*)d_in[2];   // [G,N,3]
  const int*   ei    = (const int*)  d_in[3];   // [G,2,E]
  const float* Wz    = (const float*)d_in[4];   // [3,C]
  const float* Wrad1 = (const float*)d_in[5];   // [8,C,3]
  const float* Wrad2 = (const float*)d_in[6];   // [8,C,3]
  const float* Wp    = (const float*)d_in[7];   // [7,C]
  const float* Wr0_1 = (const float*)d_in[8];
  const float* Wr2_1 = (const float*)d_in[9];
  const float* Wm0   = (const float*)d_in[10];  // [C,C]
  const float* Wr0_2 = (const float*)d_in[11];
  const float* Wr2_2 = (const float*)d_in[12];
  float* out = (float*)d_out;                   // [G,3,3]

  float* Abuf  = (float*)d_ws;                          // N*C*9
  float* hbuf  = Abuf + (size_t)N_NODES_ * C_ * 9;      // N*C
  float* fbuf  = hbuf + (size_t)N_NODES_ * C_;          // N*C
  float* B0buf = fbuf + (size_t)N_NODES_ * C_;          // N*C
  if (ws_size < (size_t)(N_NODES_ * C_ * 9 + 3 * N_NODES_ * C_) * sizeof(float))
    return;

  const int ACNT = N_NODES_ * C_ * 9;
  zero_kernel<<<1, 64, 0, stream>>>(out, G_ * 9);

  for (int g = 0; g < G_; ++g) {
    const float* rg   = x  + (size_t)g * N_EDGES_;
    const float* vg   = xv + (size_t)g * N_EDGES_ * 3;
    const float* nag  = na + (size_t)g * N_NODES_ * NZ_;
    const int*   srcg = ei + (size_t)g * 2 * N_EDGES_;
    const int*   dstg = srcg + N_EDGES_;
    float* outg = out + g * 9;

    node_embed_kernel<<<(N_NODES_ * C_) / 256, 256, 0, stream>>>(nag, Wz, hbuf);

    // pass 1
    zero_kernel<<<(ACNT + 255) / 256, 256, 0, stream>>>(Abuf, ACNT);
    edge_message_kernel<<<N_EDGES_ / 128, 256, 0, stream>>>(rg, vg, srcg, dstg,
                                                            hbuf, Wrad1, Abuf);
    node_product_readout_kernel<<<N_NODES_ / 256, 256, 0, stream>>>(
        Abuf, Wp, Wr0_1, Wr2_1, B0buf, outg);
    scalar_update_gemm_kernel<<<((N_NODES_ / 16) * 2) / 8, 256, 0, stream>>>(
        B0buf, Wm0, fbuf);

    // pass 2
    zero_kernel<<<(ACNT + 255) / 256, 256, 0, stream>>>(Abuf, ACNT);
    edge_message_kernel<<<N_EDGES_ / 128, 256, 0, stream>>>(rg, vg, srcg, dstg,
                                                            fbuf, Wrad2, Abuf);
    node_product_readout_kernel<<<N_NODES_ / 256, 256, 0, stream>>>(
        Abuf, Wp, Wr0_2, Wr2_2, B0buf, outg);
  }
}